// SparseDropout_31275951849543
// MI455X (gfx1250) — compile-verified
//
#include <hip/hip_runtime.h>

// SparseDropout: out[i] = (noise[i] >= 0.5f) ? inputs[i] * 2.0f : 0.0f
// 67.1M f32 elements, 805 MB total traffic -> ~34.5 us floor @ 23.3 TB/s.
// Main kernel is guard-free (launched only over the tile-divisible prefix):
// straight-line 4x global_load_b128 NT -> wait -> VALU -> 2x global_store_b128 NT.
// 32-bit element offsets (buffers are 268 MB) enable saddr+voffset addressing.
// Tail kernel (scalar, bounds-checked) only launches when n % TILE != 0.

typedef float v4f __attribute__((ext_vector_type(4)));

#define BLOCK 256
#define V4_PER_THREAD 2
#define TILE_FLOATS (BLOCK * V4_PER_THREAD * 4)  // 2048 floats per block

__global__ __launch_bounds__(BLOCK) void sparse_dropout_main(
    const v4f* __restrict__ vin,
    const v4f* __restrict__ vns,
    v4f* __restrict__ vout) {
  // Unsigned 32-bit indexing: max offset = nvec quads < 2^32 bytes / 16.
  const unsigned i = blockIdx.x * (BLOCK * V4_PER_THREAD) + threadIdx.x;
  const unsigned j = i + BLOCK;

  // All four 128-bit NT loads issued back-to-back: 4 outstanding LOADcnt
  // entries per thread, single s_wait_loadcnt before the VALU block.
  v4f a  = __builtin_nontemporal_load(vin + i);
  v4f na = __builtin_nontemporal_load(vns + i);
  v4f b  = __builtin_nontemporal_load(vin + j);
  v4f nb = __builtin_nontemporal_load(vns + j);

  v4f ra, rb;
#pragma unroll
  for (int k = 0; k < 4; ++k) ra[k] = (na[k] >= 0.5f) ? a[k] * 2.0f : 0.0f;
#pragma unroll
  for (int k = 0; k < 4; ++k) rb[k] = (nb[k] >= 0.5f) ? b[k] * 2.0f : 0.0f;

  __builtin_nontemporal_store(ra, vout + i);
  __builtin_nontemporal_store(rb, vout + j);
}

__global__ __launch_bounds__(BLOCK) void sparse_dropout_tail(
    const float* __restrict__ in,
    const float* __restrict__ noise,
    float* __restrict__ out,
    long long start, long long n) {
  long long t = start + blockIdx.x * (long long)BLOCK + threadIdx.x;
  if (t < n) {
    out[t] = (noise[t] >= 0.5f) ? in[t] * 2.0f : 0.0f;
  }
}

extern "C" void kernel_launch(void* const* d_in, const int* in_sizes, int n_in,
                              void* d_out, int out_size, void* d_ws, size_t ws_size,
                              hipStream_t stream) {
  const float* in    = (const float*)d_in[0];
  const float* noise = (const float*)d_in[1];
  float* out         = (float*)d_out;

  const long long n = (long long)in_sizes[0];  // 16384*4096 = 67,108,864
  const long long mainBlocks = n / TILE_FLOATS;       // 32768 for this shape
  const long long mainFloats = mainBlocks * TILE_FLOATS;

  if (mainBlocks > 0) {
    sparse_dropout_main<<<dim3((unsigned)mainBlocks), dim3(BLOCK), 0, stream>>>(
        (const v4f*)in, (const v4f*)noise, (v4f*)out);
  }

  const long long rem = n - mainFloats;  // 0 for this shape
  if (rem > 0) {
    const unsigned tailBlocks = (unsigned)((rem + BLOCK - 1) / BLOCK);
    sparse_dropout_tail<<<dim3(tailBlocks), dim3(BLOCK), 0, stream>>>(
        in, noise, out, mainFloats, n);
  }
}